// Attention_38285338476585
// MI455X (gfx1250) — compile-verified
//
#include <hip/hip_runtime.h>
#include <hip/hip_bf16.h>

typedef __attribute__((ext_vector_type(2))) float v2f;
typedef __attribute__((ext_vector_type(8))) float v8f;

#define B_DIM  32
#define S_DIM  4096
#define H_DIM  1024
#define NCHUNK 8
#define CHUNK  512              // s-rows per workgroup
#define TILE   32               // s-rows per sub-tile (128 KB of enc -> L2 hot)

// ---------------------------------------------------------------------------
// Kernel 1: q = dec(32x1024) @ W(1024x1024) using native f32 WMMA 16x16x4.
// One wave per 16x16 output tile; grid = (N/16=64, M/16=2).
// ---------------------------------------------------------------------------
__global__ void __launch_bounds__(32)
attn_proj_wmma(const float* __restrict__ dec, const float* __restrict__ W,
               float* __restrict__ q) {
    const int lane  = threadIdx.x & 31;
    const int ntile = blockIdx.x;          // 0..63
    const int mtile = blockIdx.y;          // 0..1
    const int m     = mtile * 16 + (lane & 15);
    const int n     = ntile * 16 + (lane & 15);
    const int khalf = (lane >> 4) * 2;     // 0 or 2

    v8f c = {};
    const float* arow = dec + (size_t)m * H_DIM;
    #pragma unroll 4
    for (int k0 = 0; k0 < H_DIM; k0 += 4) {
        v2f a, bm;
        a.x  = arow[k0 + khalf];
        a.y  = arow[k0 + khalf + 1];
        bm.x = W[(size_t)(k0 + khalf) * H_DIM + n];
        bm.y = W[(size_t)(k0 + khalf + 1) * H_DIM + n];
        c = __builtin_amdgcn_wmma_f32_16x16x4_f32(
                false, a, false, bm, (short)0, c, false, false);
    }

    const int row0 = mtile * 16 + (lane >> 4) * 8;
    const int col  = ntile * 16 + (lane & 15);
    #pragma unroll
    for (int r = 0; r < 8; ++r)
        q[(size_t)(row0 + r) * H_DIM + col] = c[r];
}

// ---------------------------------------------------------------------------
// Kernel 2 (fused, single pass over enc): per (batch, 512-row chunk):
//   - sub-tiles of 32 rows: wave-per-row dot products (scores), raw scores
//     to d_out prob region + LDS; online-softmax rescale; weighted float4
//     accumulation (sub-tile re-read is L2-hot).
//   - emits partial (m, l, acc[H]) per chunk.
// Grid: (NCHUNK, B), block 256.
// ---------------------------------------------------------------------------
__global__ void __launch_bounds__(256)
attn_fused(const float* __restrict__ q, const float* __restrict__ enc,
           float* __restrict__ raw_scores,   // d_out prob region (B x S)
           float* __restrict__ part_acc,     // [B][NCHUNK][H]
           float* __restrict__ part_ml) {    // [B][NCHUNK][2]
    const int b       = blockIdx.y;
    const int chunk   = blockIdx.x;
    const int s_chunk = chunk * CHUNK;

    __shared__ float qs[H_DIM];    // 4 KB
    __shared__ float ts[TILE];     // sub-tile scores

    for (int i = threadIdx.x; i < H_DIM; i += 256)
        qs[i] = q[(size_t)b * H_DIM + i];

    const int wave = threadIdx.x >> 5;
    const int lane = threadIdx.x & 31;
    const int h4   = threadIdx.x;             // owns H floats 4*h4..4*h4+3
    const float4* qv = (const float4*)qs;

    float4 acc  = {0.f, 0.f, 0.f, 0.f};
    float run_m = -INFINITY;
    float run_l = 0.f;

    __syncthreads();

    for (int t = 0; t < CHUNK / TILE; ++t) {
        const int s_base = s_chunk + t * TILE;

        // --- score phase: 8 waves x 4 rows ---
        for (int j = 0; j < TILE / 8; ++j) {
            const int r = wave * (TILE / 8) + j;       // 0..31
            const int s = s_base + r;
            const float4* row =
                (const float4*)(enc + ((size_t)b * S_DIM + s) * H_DIM);
            float d = 0.f;
            #pragma unroll
            for (int i = 0; i < 8; ++i) {
                const int idx = lane + 32 * i;
                float4 e  = row[idx];
                float4 qq = qv[idx];
                d += e.x * qq.x + e.y * qq.y + e.z * qq.z + e.w * qq.w;
            }
            #pragma unroll
            for (int off = 16; off; off >>= 1)
                d += __shfl_xor(d, off, 32);
            if (lane == 0) {
                ts[r] = d;
                raw_scores[(size_t)b * S_DIM + s] = d;
            }
        }
        __syncthreads();

        // --- online softmax update (redundant per thread, identical) ---
        float tm = -INFINITY;
        #pragma unroll
        for (int i = 0; i < TILE; ++i) tm = fmaxf(tm, ts[i]);
        const float new_m = fmaxf(run_m, tm);
        const float scale = expf(run_m - new_m);
        acc.x *= scale; acc.y *= scale; acc.z *= scale; acc.w *= scale;
        run_l *= scale;

        // --- weighted accumulation (rows are L2-hot from score phase) ---
        const float4* base =
            (const float4*)(enc + ((size_t)b * S_DIM + s_base) * H_DIM) + h4;
        #pragma unroll 4
        for (int i = 0; i < TILE; ++i) {
            const float w = expf(ts[i] - new_m);
            run_l += w;
            float4 e = base[(size_t)i * (H_DIM / 4)];
            acc.x += w * e.x; acc.y += w * e.y;
            acc.z += w * e.z; acc.w += w * e.w;
        }
        run_m = new_m;
        __syncthreads();
    }

    float4* pa = (float4*)(part_acc + ((size_t)b * NCHUNK + chunk) * H_DIM);
    pa[h4] = acc;
    if (threadIdx.x == 0) {
        part_ml[((size_t)b * NCHUNK + chunk) * 2 + 0] = run_m;
        part_ml[((size_t)b * NCHUNK + chunk) * 2 + 1] = run_l;
    }
}

// ---------------------------------------------------------------------------
// Kernel 3: per-batch finalize: merge NCHUNK partials, write attn, normalize
// raw scores into prob in place. Grid: B blocks of 256.
// ---------------------------------------------------------------------------
__global__ void __launch_bounds__(256)
attn_finalize(const float* __restrict__ part_acc,
              const float* __restrict__ part_ml,
              float* __restrict__ out) {
    const int b = blockIdx.x;
    __shared__ float sm[NCHUNK], sl[NCHUNK];
    if (threadIdx.x < NCHUNK) {
        sm[threadIdx.x] = part_ml[((size_t)b * NCHUNK + threadIdx.x) * 2 + 0];
        sl[threadIdx.x] = part_ml[((size_t)b * NCHUNK + threadIdx.x) * 2 + 1];
    }
    __syncthreads();

    float M = -INFINITY;
    #pragma unroll
    for (int c = 0; c < NCHUNK; ++c) M = fmaxf(M, sm[c]);
    float L = 0.f;
    #pragma unroll
    for (int c = 0; c < NCHUNK; ++c) L += sl[c] * expf(sm[c] - M);
    const float invL = 1.0f / L;

    // attn = sum_c acc_c * e^{m_c - M} / L
    float* attn = out + (size_t)b * H_DIM;
    for (int h = threadIdx.x; h < H_DIM; h += 256) {
        float a = 0.f;
        #pragma unroll
        for (int c = 0; c < NCHUNK; ++c)
            a += part_acc[((size_t)b * NCHUNK + c) * H_DIM + h] *
                 expf(sm[c] - M);
        attn[h] = a * invL;
    }

    // prob = e^{score - M} / L   (in place over raw scores)
    float* prob = out + (size_t)B_DIM * H_DIM + (size_t)b * S_DIM;
    for (int i = threadIdx.x; i < S_DIM; i += 256)
        prob[i] = expf(prob[i] - M) * invL;
}

// ---------------------------------------------------------------------------
extern "C" void kernel_launch(void* const* d_in, const int* in_sizes, int n_in,
                              void* d_out, int out_size, void* d_ws, size_t ws_size,
                              hipStream_t stream) {
    const float* dec = (const float*)d_in[0];   // (32, 1, 1024)
    const float* enc = (const float*)d_in[1];   // (32, 4096, 1024)
    const float* W   = (const float*)d_in[2];   // (1024, 1024)
    float* out = (float*)d_out;                 // attn (32*1024) ++ prob (32*4096)

    float* q        = (float*)d_ws;                         // 32*1024
    float* part_acc = q + (size_t)B_DIM * H_DIM;            // 32*8*1024
    float* part_ml  = part_acc + (size_t)B_DIM * NCHUNK * H_DIM;  // 32*8*2

    // 1) q = dec @ W  (f32 WMMA)
    attn_proj_wmma<<<dim3(64, 2), 32, 0, stream>>>(dec, W, q);

    // 2) fused scores + online-softmax weighted accumulation (enc read once)
    float* prob_region = out + (size_t)B_DIM * H_DIM;
    attn_fused<<<dim3(NCHUNK, B_DIM), 256, 0, stream>>>(
        q, enc, prob_region, part_acc, part_ml);

    // 3) merge partials -> attn, normalize prob in place
    attn_finalize<<<B_DIM, 256, 0, stream>>>(part_acc, part_ml, out);
}